// hybridSim_82154134438340
// MI455X (gfx1250) — compile-verified
//
#include <hip/hip_runtime.h>

#define STEPS 50
#define BLOCK 128
#define ACT_PER_BLK (BLOCK * STEPS * 2)   // floats staged per block = 12800 (51.2 KB)

#ifndef __has_builtin
#define __has_builtin(x) 0
#endif

#if __has_builtin(__builtin_amdgcn_global_load_async_to_lds_b128) && \
    __has_builtin(__builtin_amdgcn_s_wait_asynccnt)
#define USE_ASYNC_LDS 1
#else
#define USE_ASYNC_LDS 0
#endif

typedef int v4i __attribute__((ext_vector_type(4)));
typedef __attribute__((address_space(1))) v4i Gv4i;   // global (AS1) int4
typedef __attribute__((address_space(3))) v4i Lv4i;   // LDS    (AS3) int4

__global__ __launch_bounds__(BLOCK)
void hybrid_dyn_kernel(const float* __restrict__ full_states,   // (B, 1, 8)
                       const float* __restrict__ actions,       // (B, 50, 2)
                       const float* __restrict__ p_m,
                       const float* __restrict__ p_Caf,
                       const float* __restrict__ p_Car,
                       const float* __restrict__ p_lf,
                       const float* __restrict__ p_lr,
                       const float* __restrict__ p_Iz,
                       float* __restrict__ out,                 // (B, 50, 8)
                       int batch)
{
    __shared__ float s_act[ACT_PER_BLK];

    const int tid = threadIdx.x;
    const int b   = blockIdx.x * BLOCK + tid;

    // ---- Stage this block's actions (contiguous 51.2 KB) into LDS ----
    const size_t blk_base  = (size_t)blockIdx.x * ACT_PER_BLK;   // float index
    const size_t act_total = (size_t)batch * STEPS * 2;
    const float* act_blk   = actions + blk_base;

#if USE_ASYNC_LDS
    if (blk_base + ACT_PER_BLK <= act_total) {
        // Fast path (every block when BLOCK | batch): 25 unguarded async copies.
        #pragma unroll
        for (int i = 0; i < ACT_PER_BLK / (4 * BLOCK); ++i) {
            const int f4 = i * BLOCK + tid;                      // float4 slot
            __builtin_amdgcn_global_load_async_to_lds_b128(
                (Gv4i*)(act_blk + 4 * (size_t)f4),
                (Lv4i*)(s_act + 4 * f4),
                /*offset=*/0, /*cpol=*/0);
        }
    } else {
        // Tail block (only if BLOCK does not divide batch): per-lane guard.
        for (int i = 0; i < ACT_PER_BLK / (4 * BLOCK); ++i) {
            const int f4 = i * BLOCK + tid;
            if (blk_base + 4 * (size_t)f4 + 3 < act_total) {
                __builtin_amdgcn_global_load_async_to_lds_b128(
                    (Gv4i*)(act_blk + 4 * (size_t)f4),
                    (Lv4i*)(s_act + 4 * f4),
                    /*offset=*/0, /*cpol=*/0);
            }
        }
    }
    __builtin_amdgcn_s_wait_asynccnt(0);
    __syncthreads();
#else
    if (blk_base + ACT_PER_BLK <= act_total) {
        #pragma unroll
        for (int i = 0; i < ACT_PER_BLK / (4 * BLOCK); ++i) {
            const int f4 = i * BLOCK + tid;
            reinterpret_cast<float4*>(s_act)[f4] =
                reinterpret_cast<const float4*>(act_blk)[f4];
        }
    } else {
        for (int i = 0; i < ACT_PER_BLK / (4 * BLOCK); ++i) {
            const int f4 = i * BLOCK + tid;
            if (blk_base + 4 * (size_t)f4 + 3 < act_total) {
                reinterpret_cast<float4*>(s_act)[f4] =
                    reinterpret_cast<const float4*>(act_blk)[f4];
            }
        }
    }
    __syncthreads();
#endif

    if (b >= batch) return;

    // ---- Uniform scalar parameters & precomputed coefficients ----
    const float m   = *p_m;
    const float Caf = *p_Caf;
    const float Car = *p_Car;
    const float lf  = *p_lf;
    const float lr  = *p_lr;
    const float Iz  = *p_Iz;

    const float n33 = -(2.f * Caf + 2.f * Car) / m;              // A[3,3] * Vx
    const float n35 = -(2.f * Caf * lf - 2.f * Car * lr) / m;    // (A[3,5]+Vx) * Vx
    const float n53 = -(2.f * lf * Caf - 2.f * lr * Car) / Iz;   // A[5,3] * Vx
    const float n54 = -(2.f * lf * lf * Caf + 2.f * lr * lr * Car) / Iz; // A[5,4]*Vx
    const float B31 = 2.f * Caf / m;
    const float B51 = 2.f * lf * Caf / Iz;
    const float dt  = 0.01f;

    // ---- Load initial state (8 contiguous floats) ----
    float s0, s1, s2, s3, s4, s5, u0, u1;
    {
        const float4* fsp = reinterpret_cast<const float4*>(full_states) + (size_t)b * 2;
        const float4 a = fsp[0];
        const float4 c = fsp[1];
        s0 = a.x; s1 = a.y; s2 = a.z; s3 = a.w;
        s4 = c.x; s5 = c.y; u0 = c.z; u1 = c.w;
    }

    float* outp = out + (size_t)b * STEPS * 8;
    const float* my_act = s_act + tid * (STEPS * 2);

    // ---- 50-step sequential scan, state held in registers ----
    for (int k = 0; k < STEPS; ++k) {
        const float psi = s4;
        float sn, c;
        sincosf(psi, &sn, &c);

        // s_body = R(-psi) s ; only components feeding A are needed:
        const float Vx  = c * s1 + sn * s3;      // sb1 (== longitudinal speed)
        const float sb3 = c * s3 - sn * s1;      // sb3

        const float rV  = 1.0f / Vx;
        const float A33 = n33 * rV;
        const float A35 = -Vx + n35 * rV;
        const float A53 = n53 * rV;
        const float A54 = n54 * rV;

        // temp = A s_body + B u  (sparse expansion)
        const float t0 = Vx;                                     // A[0,1]=1
        const float t1 = u0;                                     // B[1,0]=1
        const float t2 = sb3;                                    // A[2,3]=1
        const float t3 = A33 * sb3 + A35 * s5 + B31 * u1;
        const float t5 = A53 * sb3 + A54 * psi + B51 * u1;
        //    t4 = sb5 = s5                                      // A[4,5]=1

        // pred = s + dt * R(psi) temp
        const float p0 = s0 + dt * (c * t0 - sn * t2);
        const float p1 = s1 + dt * (c * t1 - sn * t3);
        const float p2 = s2 + dt * (sn * t0 + c * t2);
        const float p3 = s3 + dt * (sn * t1 + c * t3);
        const float p4 = s4 + dt * s5;
        const float p5 = s5 + dt * t5;

        const float a0 = my_act[2 * k];
        const float a1 = my_act[2 * k + 1];

        float4* op = reinterpret_cast<float4*>(outp + k * 8);
        op[0] = make_float4(p0, p1, p2, p3);
        op[1] = make_float4(p4, p5, a0, a1);

        s0 = p0; s1 = p1; s2 = p2; s3 = p3; s4 = p4; s5 = p5;
        u0 = a0; u1 = a1;
    }
}

extern "C" void kernel_launch(void* const* d_in, const int* in_sizes, int n_in,
                              void* d_out, int out_size, void* d_ws, size_t ws_size,
                              hipStream_t stream) {
    (void)n_in; (void)out_size; (void)d_ws; (void)ws_size;
    const float* full_states = (const float*)d_in[0];
    const float* actions     = (const float*)d_in[1];
    const float* p_m   = (const float*)d_in[2];
    const float* p_Caf = (const float*)d_in[3];
    const float* p_Car = (const float*)d_in[4];
    const float* p_lf  = (const float*)d_in[5];
    const float* p_lr  = (const float*)d_in[6];
    const float* p_Iz  = (const float*)d_in[7];
    float* out = (float*)d_out;

    const int batch = in_sizes[0] / 8;            // full_states is (B,1,8)
    const int grid  = (batch + BLOCK - 1) / BLOCK;

    hybrid_dyn_kernel<<<grid, BLOCK, 0, stream>>>(
        full_states, actions, p_m, p_Caf, p_Car, p_lf, p_lr, p_Iz, out, batch);
}